// ReluLayer_59004260712750
// MI455X (gfx1250) — compile-verified
//
#include <hip/hip_runtime.h>
#include <stdint.h>

// Zonotope ReLU transformer for MI455X (gfx1250, wave32).
// Memory-bound (~0.2 FLOP/byte -> ~23us floor at 23.3 TB/s): optimize data movement.
//  - async global->LDS double-buffered staging (global_load_async_to_lds_b128 + s_wait_asynccnt)
//  - b128-per-lane accesses everywhere (512B per wave32 instruction)
//  - non-temporal b128 stores for the ~404MB output stream (preserve pd in L2 for pass 2)
//  - one block per (batch, column-half); each lane owns 4 consecutive columns end-to-end

typedef float v4f __attribute__((ext_vector_type(4)));

#define EPSF 1e-8f

constexpr int B_ = 256;   // batch
constexpr int N_ = 256;   // deviation terms
constexpr int D_ = 512;   // feature dim
constexpr int HALF_ = 256;     // columns handled per block
constexpr int TPB = 64;        // threads per block (2 waves, wave32)
constexpr int CPT = 4;         // columns per thread (float4 lane width)
constexpr int TILE_ROWS = 16;                       // n-rows per LDS tile
constexpr int TILE_FLOATS = TILE_ROWS * HALF_;      // 4096 floats = 16KB
constexpr int NTILES = N_ / TILE_ROWS;              // 16
constexpr int ISSUES = TILE_FLOATS / (TPB * 4);     // 16 b128 async loads / thread / tile

static_assert(TPB * CPT == HALF_, "column coverage");
static_assert(ISSUES * TPB * 4 == TILE_FLOATS, "tile coverage");

// LDS byte address (address-space-3 offset) of a generic pointer into __shared__.
__device__ __forceinline__ uint32_t lds_addr_of(const void* p) {
  return (uint32_t)(uintptr_t)(__attribute__((address_space(3))) const void*)p;
}

// GLOBAL_LOAD_ASYNC_TO_LDS_B128: copies 16B per lane from global memory to the LDS address in vdst; tracked by ASYNCcnt.
__device__ __forceinline__ void async_b128(uint32_t lds_off, const float* g) {
  asm volatile("global_load_async_to_lds_b128 %0, %1, off"
               :: "v"(lds_off), "v"((uint64_t)(uintptr_t)g)
               : "memory");
}

__global__ __launch_bounds__(TPB) void zonotope_relu_kernel(
    const float* __restrict__ cen,   // (B, D)
    const float* __restrict__ pd,    // (B, N, D)
    const float* __restrict__ err,   // (B, D)
    float* __restrict__ out_c,       // (B, D)
    float* __restrict__ out_pd,      // (B, N+D, D)
    float* __restrict__ out_e)       // (B, D)
{
  __shared__ __align__(16) float tiles[2][TILE_FLOATS];

  const int b  = blockIdx.x;
  const int h  = blockIdx.y;           // which 256-column half
  const int t  = threadIdx.x;
  const int lc = CPT * t;              // local column (0..252), thread owns lc..lc+3
  const int c0 = h * HALF_ + lc;       // global column

  const float* pdB = pd + (size_t)b * N_ * D_ + (size_t)h * HALF_;

  auto issue_tile = [&](int k, int buf) {
    const float* src = pdB + (size_t)k * TILE_ROWS * D_;
    const uint32_t lbase = lds_addr_of(&tiles[buf][0]);
#pragma unroll
    for (int i = 0; i < ISSUES; ++i) {
      const int flat = (i * TPB + t) * 4;       // float index within tile
      const int r = flat >> 8;                  // / HALF_
      const int c = flat & (HALF_ - 1);
      async_b128(lbase + (uint32_t)flat * 4u, src + (size_t)r * D_ + c);
    }
  };

  // ---- Pass 1: radius reduction over n, staged through LDS (double buffered)
  issue_tile(0, 0);
  float acc[CPT] = {0.0f, 0.0f, 0.0f, 0.0f};
  for (int k = 0; k < NTILES; ++k) {
    const int cur = k & 1;
    if (k + 1 < NTILES) {
      issue_tile(k + 1, cur ^ 1);
      // tile k's 16 loads complete when <= 16 (tile k+1's) remain; done is in-order
      asm volatile("s_wait_asynccnt 16" ::: "memory");
    } else {
      asm volatile("s_wait_asynccnt 0" ::: "memory");
    }
    __syncthreads();
    const float* buf = &tiles[cur][0];
#pragma unroll
    for (int r = 0; r < TILE_ROWS; ++r) {
      v4f v = *(const v4f*)(buf + r * HALF_ + lc);   // ds_load_b128, conflict-free
      acc[0] += __builtin_fabsf(v.x);
      acc[1] += __builtin_fabsf(v.y);
      acc[2] += __builtin_fabsf(v.z);
      acc[3] += __builtin_fabsf(v.w);
    }
    __syncthreads();  // all reads done before this buffer is overwritten
  }

  // ---- Per-column lam/beta/delta
  const size_t cIdx = (size_t)b * D_ + c0;
  const v4f cv = *(const v4f*)(cen + cIdx);
  const v4f ev = *(const v4f*)(err + cIdx);

  float lam[CPT], dlt[CPT], ncv[CPT], nev[CPT];
#pragma unroll
  for (int i = 0; i < CPT; ++i) {
    const float c = cv[i];
    const float e = ev[i];
    const float rad = acc[i] + __builtin_fabsf(e);
    const float lo = c - rad, up = c + rad;
    const float rl = fmaxf(lo, 0.0f), ru = fmaxf(up, 0.0f);
    float l = (ru - rl) / (up - lo + EPSF);
    if (lo >= 0.0f) l = 1.0f;
    if (up < 0.0f)  l = 0.0f;
    l = fminf(fmaxf(l, 0.0f), 1.0f);
    const float beta = (rl - l * lo) * 0.5f;
    lam[i] = l;
    dlt[i] = __builtin_fabsf(beta);
    ncv[i] = l * c + beta;
    nev[i] = l * e;
  }
  {
    v4f nc = {ncv[0], ncv[1], ncv[2], ncv[3]};
    v4f ne = {nev[0], nev[1], nev[2], nev[3]};
    __builtin_nontemporal_store(nc, (v4f*)(out_c + cIdx));
    __builtin_nontemporal_store(ne, (v4f*)(out_e + cIdx));
  }

  // ---- Pass 2: scaled partial deviations (pd reread L2-resident), NT b128 stores
  float* outBase = out_pd + (size_t)b * (N_ + D_) * D_;
  {
    const float* src = pd + (size_t)b * N_ * D_ + c0;
    float* dst = outBase + c0;
#pragma unroll 8
    for (int n = 0; n < N_; ++n) {
      v4f v = *(const v4f*)(src + (size_t)n * D_);
      v4f o = { v.x * lam[0], v.y * lam[1], v.z * lam[2], v.w * lam[3] };
      __builtin_nontemporal_store(o, (v4f*)(dst + (size_t)n * D_));
    }
  }

  // ---- Pass 3: diag(delta) block — mostly-zero stream, NT b128 stores
  {
    float* dbase = outBase + (size_t)N_ * D_ + c0;
#pragma unroll 8
    for (int r = 0; r < D_; ++r) {
      v4f z = {0.0f, 0.0f, 0.0f, 0.0f};
      if (r == c0 + 0) z.x = dlt[0];
      if (r == c0 + 1) z.y = dlt[1];
      if (r == c0 + 2) z.z = dlt[2];
      if (r == c0 + 3) z.w = dlt[3];
      __builtin_nontemporal_store(z, (v4f*)(dbase + (size_t)r * D_));
    }
  }
}

extern "C" void kernel_launch(void* const* d_in, const int* in_sizes, int n_in,
                              void* d_out, int out_size, void* d_ws, size_t ws_size,
                              hipStream_t stream) {
  (void)in_sizes; (void)n_in; (void)out_size; (void)d_ws; (void)ws_size;
  const float* cen = (const float*)d_in[0];   // (256, 512)
  const float* pdv = (const float*)d_in[1];   // (256, 256, 512)
  const float* er  = (const float*)d_in[2];   // (256, 512)

  float* out_c  = (float*)d_out;                              // (256, 512)
  float* out_pd = out_c + (size_t)B_ * D_;                    // (256, 768, 512)
  float* out_e  = out_pd + (size_t)B_ * (N_ + D_) * D_;       // (256, 512)

  dim3 grid(B_, D_ / HALF_);   // 256 x 2 = 512 blocks, 64 threads each
  zonotope_relu_kernel<<<grid, TPB, 0, stream>>>(cen, pdv, er, out_c, out_pd, out_e);
}